// Contracter_47622597378581
// MI455X (gfx1250) — compile-verified
//
#include <hip/hip_runtime.h>
#include <cstdint>

typedef float v2f __attribute__((ext_vector_type(2)));
typedef float v8f __attribute__((ext_vector_type(8)));

#define SCATTER_FACTOR 0.1414f

// ---- Path-weighted Wigner-3j nonzero table ---------------------------------
// Basis: index 0 = l=0; 1..3 = l=1 (y,z,x); 4..8 = l=2 (xy,yz,z2,zx,x2-y2).
// Entry: W3J[i][j][k][p] = c  ->  ww3j[i][j][k] = c * weights[p]
struct W3JEntry { unsigned char i, j, k, p; float c; };

#define C1 0.57735027f   // 1/sqrt(3)
#define C2 0.44721360f   // 1/sqrt(5)
#define A1 0.54772256f   // sqrt(3/10)
#define A2 0.31622777f   // sqrt(1/10)
#define A3 0.63245553f   // sqrt(2/5)
#define B1 0.70710678f   // sqrt(1/2)
#define B2 0.40824829f   // sqrt(1/6)
#define B3 0.81649658f   // sqrt(2/3)
#define D1 0.53452248f   // sqrt(2/7)
#define D2 0.26726124f   // sqrt(1/14)
#define D3 0.46291005f   // sqrt(3/14)

__constant__ W3JEntry g_w3j[83] = {
    // p0: (0e,0e)->0e
    {0,0,0,0, 1.0f},
    // p1: (1o,1o)->0e
    {1,1,0,1, C1},{2,2,0,1, C1},{3,3,0,1, C1},
    // p2: (2e,2e)->0e
    {4,4,0,2, C2},{5,5,0,2, C2},{6,6,0,2, C2},{7,7,0,2, C2},{8,8,0,2, C2},
    // p3: (0e,1o)->1o
    {0,1,1,3, 1.0f},{0,2,2,3, 1.0f},{0,3,3,3, 1.0f},
    // p4: (1o,0e)->1o
    {1,0,1,4, 1.0f},{2,0,2,4, 1.0f},{3,0,3,4, 1.0f},
    // p5: (1o,2e)->1o
    {3,4,1,5, A1},{1,4,3,5, A1},
    {1,5,2,5, A1},{2,5,1,5, A1},
    {1,6,1,5,-A2},{2,6,2,5, A3},{3,6,3,5,-A2},
    {2,7,3,5, A1},{3,7,2,5, A1},
    {1,8,1,5,-A1},{3,8,3,5, A1},
    // p6: (2e,1o)->1o
    {4,3,1,6, A1},{4,1,3,6, A1},
    {5,1,2,6, A1},{5,2,1,6, A1},
    {6,1,1,6,-A2},{6,2,2,6, A3},{6,3,3,6,-A2},
    {7,2,3,6, A1},{7,3,2,6, A1},
    {8,1,1,6,-A1},{8,3,3,6, A1},
    // p7: (0e,2e)->2e
    {0,4,4,7, 1.0f},{0,5,5,7, 1.0f},{0,6,6,7, 1.0f},{0,7,7,7, 1.0f},{0,8,8,7, 1.0f},
    // p8: (1o,1o)->2e
    {3,1,4,8, B1},{1,3,4,8, B1},
    {1,2,5,8, B1},{2,1,5,8, B1},
    {1,1,6,8,-B2},{2,2,6,8, B3},{3,3,6,8,-B2},
    {2,3,7,8, B1},{3,2,7,8, B1},
    {1,1,8,8,-B1},{3,3,8,8, B1},
    // p9: (2e,0e)->2e
    {4,0,4,9, 1.0f},{5,0,5,9, 1.0f},{6,0,6,9, 1.0f},{7,0,7,9, 1.0f},{8,0,8,9, 1.0f},
    // p10: (2e,2e)->2e
    {6,6,6,10, D1},
    {4,4,6,10,-D1},{4,6,4,10,-D1},{6,4,4,10,-D1},
    {6,8,8,10,-D1},{8,6,8,10,-D1},{8,8,6,10,-D1},
    {5,5,6,10, D2},{5,6,5,10, D2},{6,5,5,10, D2},
    {7,7,6,10, D2},{7,6,7,10, D2},{6,7,7,10, D2},
    {5,5,8,10,-D3},{5,8,5,10,-D3},{8,5,5,10,-D3},
    {7,7,8,10, D3},{7,8,7,10, D3},{8,7,7,10, D3},
    {4,5,7,10, D3},{4,7,5,10, D3},{5,4,7,10, D3},
    {5,7,4,10, D3},{7,4,5,10, D3},{7,5,4,10, D3},
};

// ---- Kernel 1: zero the atom accumulation buffer (n_atoms read on device) --
__global__ void zero_atoms_kernel(float* __restrict__ atoms,
                                  const int* __restrict__ n_atoms_p) {
  int total = n_atoms_p[0] * 288;
  for (int t = blockIdx.x * blockDim.x + threadIdx.x; t < total;
       t += gridDim.x * blockDim.x)
    atoms[t] = 0.0f;
}

// ---- Kernel 2: scaled scatter-add of x2 into atom buffer (L2-resident) -----
__global__ void scatter_kernel(const float* __restrict__ x2,
                               const int* __restrict__ idxs,
                               float* __restrict__ atoms, long long total) {
  long long gid = (long long)blockIdx.x * blockDim.x + threadIdx.x;
  if (gid >= total) return;
  int e = (int)(gid / 288);
  int c = (int)(gid % 288);
  float v = SCATTER_FACTOR * __builtin_nontemporal_load(x2 + gid);
  unsafeAtomicAdd(atoms + (size_t)idxs[e] * 288 + c, v);
}

// ---- Kernel 3: gather + WMMA contraction -----------------------------------
// GEMM view: out[16 rows, 9] = A[16, 84] x B[84, 16], A[M,q] = x1[M,i]*x2[M,j]
// with q = i*9+j (81 real, padded to 84 = 21 chunks of K=4), B[q,n] = ww3j[i,j,n].
__global__ void __launch_bounds__(256)
contract_wmma_kernel(const float* __restrict__ x1, const int* __restrict__ idxs,
                     const float* __restrict__ atoms,
                     const float* __restrict__ weights, float* __restrict__ out,
                     int n_tiles) {
  __shared__ float Wl[84 * 16];          // padded ww3j as 84x16 B-matrix
  __shared__ float rowbuf[8][16 * 20];   // per-wave tile staging (18 used, pad 20)

  // Build path-weighted Wigner matrix once per block.
  for (int t = threadIdx.x; t < 84 * 16; t += blockDim.x) Wl[t] = 0.0f;
  __syncthreads();
  if (threadIdx.x < 83) {
    W3JEntry e = g_w3j[threadIdx.x];
    Wl[((int)e.i * 9 + (int)e.j) * 16 + (int)e.k] = e.c * weights[e.p];
  }
  __syncthreads();

  const int lane = threadIdx.x & 31;
  const int m    = lane & 15;        // row within tile (A/M) and column (B/N)
  const bool hi  = lane >= 16;       // high half-wave: K slots 2,3
  const int kk   = hi ? 2 : 0;
  const int wave_in_block = __builtin_amdgcn_readfirstlane((int)(threadIdx.x >> 5));
  const int gwave  = blockIdx.x * (blockDim.x >> 5) + wave_in_block;
  const int nwaves = gridDim.x * (blockDim.x >> 5);

  // Preload all 21 B fragments (loop-invariant) into registers.
  float bre0[21], bre1[21];
#pragma unroll
  for (int c = 0; c < 21; ++c) {
    bre0[c] = Wl[(4 * c + kk) * 16 + m];
    bre1[c] = Wl[(4 * c + kk + 1) * 16 + m];
  }

  float* mybuf = rowbuf[wave_in_block];

  for (int tile = gwave; tile < n_tiles; tile += nwaves) {
    const int row0 = tile * 16;
    // A 16-row tile is exactly half of one edge (MUL=32): idxs is wave-uniform
    // and the x2 gather region is one contiguous 144-dword block.
    const int z    = __builtin_amdgcn_readfirstlane(row0 >> 5);
    const int u0   = row0 & 31;  // 0 or 16
    const int atom = idxs[z];
    const float* x1base = x1 + (size_t)row0 * 9;
    const float* x2base = atoms + (size_t)atom * 288 + (size_t)u0 * 9;

    // Cooperative, coalesced tile load into per-wave LDS.
#pragma unroll
    for (int it = 0; it < 5; ++it) {
      int t = lane + it * 32;
      if (t < 144) {
        int r = t / 9, cc = t % 9;
        mybuf[r * 20 + cc]     = __builtin_nontemporal_load(x1base + t);
        mybuf[r * 20 + 9 + cc] = x2base[t];
      }
    }
    asm volatile("s_wait_dscnt 0" ::: "memory");  // wave-local LDS visibility

    // Each lane pulls its row's 18 values into registers.
    float x1r[9], x2r[9];
#pragma unroll
    for (int q = 0; q < 9; ++q) {
      x1r[q] = mybuf[m * 20 + q];
      x2r[q] = mybuf[m * 20 + 9 + q];
    }

    v8f acc = {0.f, 0.f, 0.f, 0.f, 0.f, 0.f, 0.f, 0.f};
#pragma unroll
    for (int c = 0; c < 21; ++c) {
      const int qlo = 4 * c, qhi = 4 * c + 2;
      // products for K slots 0,1 (low half) and 2,3 (high half); q>80 padded 0
      float plo0 = x1r[qlo / 9] * x2r[qlo % 9];
      float plo1 = (qlo + 1 <= 80) ? x1r[(qlo + 1) / 9] * x2r[(qlo + 1) % 9] : 0.0f;
      float phi0 = (qhi     <= 80) ? x1r[qhi / 9] * x2r[qhi % 9]             : 0.0f;
      float phi1 = (qhi + 1 <= 80) ? x1r[(qhi + 1) / 9] * x2r[(qhi + 1) % 9] : 0.0f;
      v2f A; A.x = hi ? phi0 : plo0; A.y = hi ? phi1 : plo1;  // cndmask, EXEC stays full
      v2f B; B.x = bre0[c]; B.y = bre1[c];
      acc = __builtin_amdgcn_wmma_f32_16x16x4_f32(false, A, false, B,
                                                  (short)0, acc, false, false);
    }

    // C/D layout: lane n = m (valid n<9); VGPR r -> M = r (+8 for high half).
    if (m < 9) {
#pragma unroll
      for (int r = 0; r < 8; ++r) {
        int M = r + (hi ? 8 : 0);
        __builtin_nontemporal_store(acc[r], out + (size_t)(row0 + M) * 9 + m);
      }
    }
  }
}

// ---- Host launcher ----------------------------------------------------------
extern "C" void kernel_launch(void* const* d_in, const int* in_sizes, int n_in,
                              void* d_out, int out_size, void* d_ws, size_t ws_size,
                              hipStream_t stream) {
  (void)n_in; (void)out_size; (void)ws_size;
  const float* x1       = (const float*)d_in[0];
  const float* x2       = (const float*)d_in[1];
  const int*   idxs     = (const int*)d_in[2];
  const int*   n_atoms  = (const int*)d_in[3];  // device-side scalar
  const float* weights  = (const float*)d_in[4];
  float*       out      = (float*)d_out;
  float*       atoms    = (float*)d_ws;         // n_atoms * 288 floats (~2.3 MB)

  const int n_edges = in_sizes[0] / 288;        // MUL(32) * 9 per edge
  const int n_rows  = n_edges * 32;             // always a multiple of 32
  const int n_tiles = n_rows / 16;              // no partial tiles

  zero_atoms_kernel<<<512, 256, 0, stream>>>(atoms, n_atoms);

  const long long total = (long long)n_edges * 288;
  const int sblocks = (int)((total + 255) / 256);
  scatter_kernel<<<sblocks, 256, 0, stream>>>(x2, idxs, atoms, total);

  contract_wmma_kernel<<<4096, 256, 0, stream>>>(x1, idxs, atoms, weights, out,
                                                 n_tiles);
}